// EdgeClassifierGNN_39316130627626
// MI455X (gfx1250) — compile-verified
//
#include <hip/hip_runtime.h>

#define NODE_IN 128
#define HIDDEN  128
#define EDGE_IN 16
#define EM_K    272   // 2*HIDDEN + EDGE_IN
#define SG_PAD  260   // 256 + 4 pad (bank-conflict-free LDS rows)
#define EM_PAD  276   // 272 + 4 pad

typedef __attribute__((ext_vector_type(2))) float v2f;
typedef __attribute__((ext_vector_type(4))) float v4f;
typedef __attribute__((ext_vector_type(8))) float v8f;

__device__ __forceinline__ v8f wmma4(v2f a, v2f b, v8f c) {
    // D(16x16 f32) = A(16x4 f32) * B(4x16 f32) + C   -> v_wmma_f32_16x16x4_f32
    return __builtin_amdgcn_wmma_f32_16x16x4_f32(false, a, false, b, (short)0, c,
                                                 false, false);
}

// ---------------------------------------------------------------- zero scratch
__global__ void zero_kernel(float* __restrict__ p, int n) {
    int i = blockIdx.x * blockDim.x + threadIdx.x;
    if (i < n) p[i] = 0.0f;
}

// ------------------------------------------------- scatter-sum + degree (atomics)
// one wave per edge: 32 lanes x 4 features; atomics resolve in L2 (buffer fits)
__global__ void scatter_kernel(const float* __restrict__ h,
                               const int* __restrict__ src,
                               const int* __restrict__ dst,
                               float* __restrict__ msg,
                               float* __restrict__ deg, int n_edges) {
    int lane = threadIdx.x & 31;
    int e = blockIdx.x * (blockDim.x >> 5) + (threadIdx.x >> 5);
    if (e >= n_edges) return;
    int s = src[e];
    int d = dst[e];
    const float* hr = h + (size_t)s * HIDDEN + lane * 4;
    float* mr = msg + (size_t)d * HIDDEN + lane * 4;
    v4f v = *(const v4f*)hr;
    unsafeAtomicAdd(mr + 0, v.x);
    unsafeAtomicAdd(mr + 1, v.y);
    unsafeAtomicAdd(mr + 2, v.z);
    unsafeAtomicAdd(mr + 3, v.w);
    if (deg != nullptr && lane == 0) unsafeAtomicAdd(deg + d, 1.0f);
}

// ------------------------------------------------------- mean = sum / max(deg,1)
__global__ void normalize_kernel(float* __restrict__ msg,
                                 const float* __restrict__ deg, int nh) {
    int i = blockIdx.x * blockDim.x + threadIdx.x;
    if (i < nh) {
        float d = deg[i >> 7];  // HIDDEN == 128
        msg[i] = msg[i] * (1.0f / fmaxf(d, 1.0f));
    }
}

// --------------------------------------------- SAGE layer: relu(mean@Wl^T + b + x@Wr^T)
// block = 256 threads = 8 waves; block owns 16 nodes, wave w owns output cols [16w,16w+16)
__global__ void sage_gemm_kernel(const float* __restrict__ mean,
                                 const float* __restrict__ xin,
                                 const float* __restrict__ Wl,
                                 const float* __restrict__ Wr,
                                 const float* __restrict__ bias,
                                 float* __restrict__ out) {
    __shared__ float As[16 * SG_PAD];  // [16 nodes][mean(128) | x(128)] padded
    const int tid = threadIdx.x;
    const int node0 = blockIdx.x * 16;
    for (int i = tid; i < 16 * 128; i += 256) {
        int r = i >> 7, c = i & 127;
        As[r * SG_PAD + c]       = mean[(size_t)(node0 + r) * 128 + c];
        As[r * SG_PAD + 128 + c] = xin[(size_t)(node0 + r) * 128 + c];
    }
    __syncthreads();

    const int wid  = tid >> 5;          // output tile N
    const int lane = tid & 31;
    const int row  = lane & 15;         // M row for A, N col for B
    const int koff = (lane >> 4) << 1;  // K sub-offset {0,2}
    const int j = wid * 16 + row;       // this lane's B column (weight row)

    const float* arow  = As + row * SG_PAD + koff;
    const float* wlrow = Wl + (size_t)j * 128 + koff;
    const float* wrrow = Wr + (size_t)j * 128 + koff;

    v8f c = {};
#pragma unroll
    for (int kk = 0; kk < 128; kk += 4) {
        v2f a = *(const v2f*)(arow + kk);
        v2f b = *(const v2f*)(wlrow + kk);
        c = wmma4(a, b, c);
    }
#pragma unroll
    for (int kk = 0; kk < 128; kk += 4) {
        v2f a = *(const v2f*)(arow + 128 + kk);
        v2f b = *(const v2f*)(wrrow + kk);
        c = wmma4(a, b, c);
    }

    const float bj = bias[j];
    const int mbase = node0 + ((lane >> 4) << 3);
#pragma unroll
    for (int v = 0; v < 8; ++v) {
        float val = fmaxf(c[v] + bj, 0.0f);
        out[(size_t)(mbase + v) * 128 + wid * 16 + row] = val;
    }
}

// ------------- edge MLP: out = relu([h[ps],h[pd],attr] @ Wm1^T + bm1) @ Wm2^T + bm2
// block = 256 threads = 8 waves; block owns 32 edges (two 16-edge M tiles),
// wave w owns hidden cols [16w,16w+16). Each global B load feeds TWO WMMAs
// (one per M tile) -> halves global-load pressure per WMMA.
__global__ void edge_mlp_kernel(const float* __restrict__ h,
                                const int* __restrict__ ps,
                                const int* __restrict__ pd,
                                const float* __restrict__ attr,
                                const float* __restrict__ Wm1,
                                const float* __restrict__ bm1,
                                const float* __restrict__ Wm2,
                                const float* __restrict__ bm2,
                                float* __restrict__ out) {
    __shared__ float As[32 * EM_PAD];   // [32 edges][272] padded
    __shared__ float partial[8][32];
    const int tid = threadIdx.x;
    const int e0 = blockIdx.x * 32;

    for (int i = tid; i < 32 * EM_K; i += 256) {
        int r = i / EM_K;
        int cidx = i - r * EM_K;
        int e = e0 + r;
        float v;
        if (cidx < 128)      v = h[(size_t)ps[e] * 128 + cidx];
        else if (cidx < 256) v = h[(size_t)pd[e] * 128 + (cidx - 128)];
        else                 v = attr[(size_t)e * EDGE_IN + (cidx - 256)];
        As[r * EM_PAD + cidx] = v;
    }
    __syncthreads();

    const int wid  = tid >> 5;
    const int lane = tid & 31;
    const int row  = lane & 15;
    const int koff = (lane >> 4) << 1;
    const int j = wid * 16 + row;  // hidden unit handled by this lane (B column)

    const float* a0   = As + row * EM_PAD + koff;         // M tile 0 (edges 0..15)
    const float* a1   = As + (16 + row) * EM_PAD + koff;  // M tile 1 (edges 16..31)
    const float* wrow = Wm1 + (size_t)j * EM_K + koff;

    v8f c0 = {};
    v8f c1 = {};
#pragma unroll
    for (int kk = 0; kk < EM_K; kk += 4) {
        v2f b  = *(const v2f*)(wrow + kk);
        v2f x0 = *(const v2f*)(a0 + kk);
        v2f x1 = *(const v2f*)(a1 + kk);
        c0 = wmma4(x0, b, c0);
        c1 = wmma4(x1, b, c1);
    }

    // fused: relu(hidden)*Wm2[j], then reduce the 16 columns across lanes
    const float bj = bm1[j];
    const float w2 = Wm2[j];
    float acc0[8], acc1[8];
#pragma unroll
    for (int v = 0; v < 8; ++v) {
        acc0[v] = fmaxf(c0[v] + bj, 0.0f) * w2;
        acc1[v] = fmaxf(c1[v] + bj, 0.0f) * w2;
    }
#pragma unroll
    for (int m = 1; m < 16; m <<= 1) {
#pragma unroll
        for (int v = 0; v < 8; ++v) {
            acc0[v] += __shfl_xor(acc0[v], m, 32);
            acc1[v] += __shfl_xor(acc1[v], m, 32);
        }
    }
    if (row == 0) {
        // lane 0 holds edges 0..7 (VGPR v -> M=v), lane 16 holds edges 8..15
        const int base = (lane >> 4) << 3;
#pragma unroll
        for (int v = 0; v < 8; ++v) {
            partial[wid][base + v]      = acc0[v];
            partial[wid][16 + base + v] = acc1[v];
        }
    }
    __syncthreads();
    if (tid < 32) {
        float s = bm2[0];
#pragma unroll
        for (int w = 0; w < 8; ++w) s += partial[w][tid];
        out[e0 + tid] = s;
    }
}

// --------------------------------------------------------------------- driver
extern "C" void kernel_launch(void* const* d_in, const int* in_sizes, int n_in,
                              void* d_out, int out_size, void* d_ws, size_t ws_size,
                              hipStream_t stream) {
    const float* x     = (const float*)d_in[0];
    const int*   mp    = (const int*)d_in[1];
    const int*   pred  = (const int*)d_in[2];
    const float* pattr = (const float*)d_in[3];
    const float* W1l   = (const float*)d_in[4];
    const float* b1    = (const float*)d_in[5];
    const float* W1r   = (const float*)d_in[6];
    const float* W2l   = (const float*)d_in[7];
    const float* b2    = (const float*)d_in[8];
    const float* W2r   = (const float*)d_in[9];
    const float* Wm1   = (const float*)d_in[10];
    const float* bm1   = (const float*)d_in[11];
    const float* Wm2   = (const float*)d_in[12];
    const float* bm2   = (const float*)d_in[13];

    const int n_nodes = in_sizes[0] / NODE_IN;   // 50000 (multiple of 16)
    const int n_mp    = in_sizes[1] / 2;         // 800000
    const int n_pred  = in_sizes[3] / EDGE_IN;   // 400000 (multiple of 32)
    const int nh      = n_nodes * HIDDEN;

    const int* mp_src = mp;
    const int* mp_dst = mp + n_mp;
    const int* p_src  = pred;
    const int* p_dst  = pred + n_pred;

    float* msg = (float*)d_ws;                   // mean computed in-place
    float* h1  = msg + (size_t)nh;
    float* h2  = h1 + (size_t)nh;
    float* deg = h2 + (size_t)nh;

    // ---- layer 1: scatter-mean(x) -> GEMM ----
    zero_kernel<<<(nh + 255) / 256, 256, 0, stream>>>(msg, nh);
    zero_kernel<<<(n_nodes + 255) / 256, 256, 0, stream>>>(deg, n_nodes);
    scatter_kernel<<<(n_mp + 7) / 8, 256, 0, stream>>>(x, mp_src, mp_dst, msg, deg, n_mp);
    normalize_kernel<<<(nh + 255) / 256, 256, 0, stream>>>(msg, deg, nh);
    sage_gemm_kernel<<<n_nodes / 16, 256, 0, stream>>>(msg, x, W1l, W1r, b1, h1);

    // ---- layer 2: scatter-mean(h1) -> GEMM (degree unchanged) ----
    zero_kernel<<<(nh + 255) / 256, 256, 0, stream>>>(msg, nh);
    scatter_kernel<<<(n_mp + 7) / 8, 256, 0, stream>>>(h1, mp_src, mp_dst, msg, nullptr, n_mp);
    normalize_kernel<<<(nh + 255) / 256, 256, 0, stream>>>(msg, deg, nh);
    sage_gemm_kernel<<<n_nodes / 16, 256, 0, stream>>>(msg, h1, W2l, W2r, b2, h2);

    // ---- edge MLP over prediction edges (32 edges / block) ----
    edge_mlp_kernel<<<n_pred / 32, 256, 0, stream>>>(h2, p_src, p_dst, pattr,
                                                     Wm1, bm1, Wm2, bm2,
                                                     (float*)d_out);
}